// _Net_85899345920420
// MI455X (gfx1250) — compile-verified
//
#include <hip/hip_runtime.h>

typedef float v2f __attribute__((ext_vector_type(2)));
typedef float v8f __attribute__((ext_vector_type(8)));

#define NF 128  // feature width for both layers

// ---------------- degree / normalization ----------------

__global__ void k_init_deg(float* deg, int n) {
    int i = blockIdx.x * blockDim.x + threadIdx.x;
    if (i < n) deg[i] = 1.0f;  // self-loop contribution
}

__global__ void k_accum_deg(const int* __restrict__ dst, int E, float* deg) {
    int e = blockIdx.x * blockDim.x + threadIdx.x;
    if (e < E) atomicAdd(&deg[dst[e]], 1.0f);
}

__global__ void k_rsqrt_inplace(float* deg, int n) {
    int i = blockIdx.x * blockDim.x + threadIdx.x;
    if (i < n) deg[i] = rsqrtf(deg[i]);  // deg >= 1 always (self-loops)
}

// ---------------- GEMM: H[N,128] = X[N,128] @ W[128,128] via f32 WMMA ----------------
// Each wave computes a 16x128 output tile: 8 accumulators of 16x16,
// K looped in steps of 4 with V_WMMA_F32_16X16X4_F32.
// A 16x4 f32 layout: lanes 0-15 hold M=lane with K=k+0,k+1; lanes 16-31 hold M=lane-16 with K=k+2,k+3.
// B 4x16 f32 layout (mirror of A): lanes 0-15 hold N=lane rows K=k+0,k+1; lanes 16-31 rows K=k+2,k+3.
// C/D 16x16 f32: VGPR r -> row r (lanes 0-15) / row r+8 (lanes 16-31), col = lane&15.

__global__ __launch_bounds__(256) void k_gemm128(const float* __restrict__ X,
                                                 const float* __restrict__ W,
                                                 float* __restrict__ H,
                                                 int nrows) {
    __shared__ float sW[NF * NF];  // 64 KB, well under 320 KB/WGP
    for (int i = threadIdx.x; i < NF * NF; i += blockDim.x) sW[i] = W[i];
    __syncthreads();

    const int wave = threadIdx.x >> 5;
    const int lane = threadIdx.x & 31;
    const int row0 = (blockIdx.x * 8 + wave) * 16;
    if (row0 >= nrows) return;  // wave-uniform exit; EXEC stays all-1s for WMMA

    const int lm = lane & 15;
    const int kh = (lane >> 4) << 1;  // 0 for lanes 0-15, 2 for lanes 16-31

    v8f acc[8];
#pragma unroll
    for (int t = 0; t < 8; ++t) acc[t] = (v8f){};

    const float* xrow = X + (size_t)(row0 + lm) * NF + kh;

    for (int k = 0; k < NF; k += 4) {
        v2f a = *(const v2f*)(xrow + k);  // 8B-aligned: kh,k even
        const int kb = k + kh;
#pragma unroll
        for (int t = 0; t < 8; ++t) {
            v2f b;
            const int c0 = t * 16 + lm;
            b.x = sW[kb * NF + c0];
            b.y = sW[(kb + 1) * NF + c0];
            acc[t] = __builtin_amdgcn_wmma_f32_16x16x4_f32(
                false, a, false, b, (short)0, acc[t], false, false);
        }
    }

    // store D: lanes 0-15 -> rows row0+r, lanes 16-31 -> rows row0+8+r
    const int rbase = row0 + ((lane >> 4) << 3);
#pragma unroll
    for (int t = 0; t < 8; ++t) {
#pragma unroll
        for (int r = 0; r < 8; ++r) {
            H[(size_t)(rbase + r) * NF + t * 16 + lm] = acc[t][r];
        }
    }
}

// ---------------- edge scatter: agg[dst] += h[src] * dinv[src]*dinv[dst] ----------------
// One wave per edge; each lane moves 4 channels (float4 gather + 4 f32 atomic adds).

__global__ __launch_bounds__(256) void k_scatter(const float4* __restrict__ H4,
                                                 const int* __restrict__ src,
                                                 const int* __restrict__ dst,
                                                 const float* __restrict__ dinv,
                                                 float* __restrict__ agg,
                                                 int E) {
    const int wave = blockIdx.x * (blockDim.x >> 5) + (threadIdx.x >> 5);
    const int lane = threadIdx.x & 31;
    if (wave >= E) return;  // wave-uniform

    const int s = src[wave];
    const int d = dst[wave];
    const float w = dinv[s] * dinv[d];
    float4 v = H4[(size_t)s * (NF / 4) + lane];
    float* out = agg + (size_t)d * NF + lane * 4;
    atomicAdd(out + 0, v.x * w);
    atomicAdd(out + 1, v.y * w);
    atomicAdd(out + 2, v.z * w);
    atomicAdd(out + 3, v.w * w);
}

// ---------------- epilogue: out = (agg + h*dinv^2 + bias), optional ReLU ----------------

__global__ void k_finish(const float* __restrict__ agg, const float* __restrict__ H,
                         const float* __restrict__ dinv, const float* __restrict__ bias,
                         float* __restrict__ out, int n_elems, int relu) {
    int i = blockIdx.x * blockDim.x + threadIdx.x;
    if (i >= n_elems) return;
    const int node = i >> 7;  // /128
    const int c = i & (NF - 1);
    const float di = dinv[node];
    float v = agg[i] + H[i] * di * di + bias[c];
    if (relu) v = fmaxf(v, 0.0f);
    out[i] = v;
}

// ---------------- driver ----------------

extern "C" void kernel_launch(void* const* d_in, const int* in_sizes, int n_in,
                              void* d_out, int out_size, void* d_ws, size_t ws_size,
                              hipStream_t stream) {
    (void)n_in; (void)out_size; (void)ws_size;

    const float* x  = (const float*)d_in[0];
    const int*   ei = (const int*)d_in[1];   // edge_index [2,E]
    const float* W1 = (const float*)d_in[2];
    const float* b1 = (const float*)d_in[3];
    const float* W2 = (const float*)d_in[4];
    const float* b2 = (const float*)d_in[5];

    const int N = in_sizes[0] / NF;  // 50000
    const int E = in_sizes[1] / 2;   // 800000
    const int* src = ei;
    const int* dst = ei + E;

    char* ws = (char*)d_ws;
    float* dinv = (float*)ws;                         // N floats (deg -> dinv in place)
    size_t off = (((size_t)N * 4) + 255) & ~(size_t)255;
    const size_t hb = (size_t)N * NF * sizeof(float); // 25.6 MB
    float* bufA = (float*)(ws + off);                 // h1, then h2
    float* bufB = (float*)(ws + off + hb);            // agg1 -> y1 (in place)
    float* out  = (float*)d_out;

    const int nelem = N * NF;

    // normalization: deg (self-loop=1) -> += edges -> rsqrt
    k_init_deg<<<(N + 255) / 256, 256, 0, stream>>>(dinv, N);
    k_accum_deg<<<(E + 255) / 256, 256, 0, stream>>>(dst, E, dinv);
    k_rsqrt_inplace<<<(N + 255) / 256, 256, 0, stream>>>(dinv, N);

    const int gemm_blocks = (N + 127) / 128;  // 128 rows per block (8 waves x 16 rows)
    const int scat_blocks = (E + 7) / 8;      // 8 edges per block (1 per wave)

    // layer 1: h1 = x@W1 ; agg1 = scatter(h1) ; y1 = relu(agg1 + h1*dinv^2 + b1)
    k_gemm128<<<gemm_blocks, 256, 0, stream>>>(x, W1, bufA, N);
    hipMemsetAsync(bufB, 0, hb, stream);
    k_scatter<<<scat_blocks, 256, 0, stream>>>((const float4*)bufA, src, dst, dinv, bufB, E);
    k_finish<<<(nelem + 255) / 256, 256, 0, stream>>>(bufB, bufA, dinv, b1, bufB, nelem, 1);

    // layer 2: h2 = y1@W2 ; out = scatter(h2) + h2*dinv^2 + b2
    k_gemm128<<<gemm_blocks, 256, 0, stream>>>(bufB, W2, bufA, N);
    hipMemsetAsync(out, 0, hb, stream);
    k_scatter<<<scat_blocks, 256, 0, stream>>>((const float4*)bufA, src, dst, dinv, out, E);
    k_finish<<<(nelem + 255) / 256, 256, 0, stream>>>(out, bufA, dinv, b2, out, nelem, 0);
}